// hqconv_ansatz_layer_5703716569533
// MI455X (gfx1250) — compile-verified
//
#include <hip/hip_runtime.h>

// CDNA5 / gfx1250: wave32, WMMA f32 16x16x4.
typedef float v2f __attribute__((ext_vector_type(2)));
typedef float v8f __attribute__((ext_vector_type(8)));

static __device__ __forceinline__ v2f mk2(float a, float b) {
    v2f r; r.x = a; r.y = b; return r;
}
static __device__ __forceinline__ v8f zero8() {
    v8f z = {0.f,0.f,0.f,0.f,0.f,0.f,0.f,0.f}; return z;
}
// D = A(16x4) * B(4x16) + C, fp32 WMMA. f32 has no A/B NEG -> pass false.
static __device__ __forceinline__ v8f wmma4(v2f a, v2f b, v8f c) {
    return __builtin_amdgcn_wmma_f32_16x16x4_f32(false, a, false, b, (short)0, c, false, false);
}

// ---------------------------------------------------------------------------
// Workspace layout (floats):
//   [b*512 .. b*512+255]      : U_b real (row-major 16x16), b = 0,1,2
//   [b*512+256 .. b*512+511]  : U_b imag
//   [1536..1543]              : T12 (re[4], im[4])  combined ctrl-(RX*RZphase)
//   [1544..1551]              : T13 (re[4], im[4])
// ---------------------------------------------------------------------------

// Build the per-block 16x16 complex unitaries (product of 4 CRZ+CRX embedded
// gates) and the two cross-block combined controlled 2x2 matrices.
__global__ __launch_bounds__(256) void build_unitaries(const float* __restrict__ W,
                                                       float* __restrict__ ws) {
    __shared__ float Ur[256];
    __shared__ float Ui[256];
    const int tid = threadIdx.x;
    const int r = tid >> 4, c = tid & 15;
    const int b = blockIdx.x;   // block 0,1,2

    Ur[tid] = (r == c) ? 1.f : 0.f;
    Ui[tid] = 0.f;
    __syncthreads();

    // local (control,target) bit positions inside 4-bit index
    // wires (4b+0,4b+3),(4b+3,4b+2),(4b+2,4b+1),(4b+1,4b+0)
    // wire w -> bit 3-(w-4b)  (wire 0 is most significant, ref axis order)
    const int CBs[4] = {3, 0, 1, 2};
    const int TBs[4] = {0, 1, 2, 3};

    for (int j = 0; j < 4; ++j) {
        const int g = 4 * b + j;
        const float cz = cosf(0.5f * W[2*g]),   sz = sinf(0.5f * W[2*g]);
        const float cx = cosf(0.5f * W[2*g+1]), sx = sinf(0.5f * W[2*g+1]);
        // Control=1 2x2 block: RX(phi) * diag(e^{-i th/2}, e^{+i th/2})
        const float Tr[4] = {  cx*cz,  sx*sz, -sx*sz,  cx*cz };
        const float Ti[4] = { -cx*sz, -sx*cz, -sx*cz,  cx*sz };

        const int CB = CBs[j], TB = TBs[j];
        const int mask = (1 << CB) | (1 << TB);

        float ar = 0.f, ai = 0.f;
        for (int k = 0; k < 16; ++k) {
            float gr = 0.f, gi = 0.f;
            if ((r & ~mask) == (k & ~mask)) {
                const int rc = (r >> CB) & 1, rt = (r >> TB) & 1;
                const int kc = (k >> CB) & 1, kt = (k >> TB) & 1;
                if (!rc) { if (!kc && kt == rt) gr = 1.f; }
                else if (kc) { gr = Tr[rt*2 + kt]; gi = Ti[rt*2 + kt]; }
            }
            const float ur = Ur[k*16 + c], ui = Ui[k*16 + c];
            ar += gr*ur - gi*ui;
            ai += gr*ui + gi*ur;
        }
        __syncthreads();
        Ur[tid] = ar; Ui[tid] = ai;
        __syncthreads();
    }
    ws[b*512 + tid]       = Ur[tid];
    ws[b*512 + 256 + tid] = Ui[tid];

    if (b == 0 && tid < 2) {            // T12 (gate 12), T13 (gate 13)
        const int g = 12 + tid;
        const float cz = cosf(0.5f * W[2*g]),   sz = sinf(0.5f * W[2*g]);
        const float cx = cosf(0.5f * W[2*g+1]), sx = sinf(0.5f * W[2*g+1]);
        float* p = ws + 1536 + tid*8;
        p[0] =  cx*cz; p[1] =  sx*sz; p[2] = -sx*sz; p[3] =  cx*cz;
        p[4] = -cx*sz; p[5] = -sx*cz; p[6] = -sx*cz; p[7] =  cx*sz;
    }
}

// ---------------------------------------------------------------------------
// One workgroup (8 wave32) per circuit. State (re/im) lives in LDS as a
// [16(h)=wires0-3][16(m)=wires4-7][16(l)=wires8-11] tensor, index i=h*256+m*16+l.
// ---------------------------------------------------------------------------
__global__ __launch_bounds__(256) void qsim(const float* __restrict__ x,
                                            const float* __restrict__ ws,
                                            float* __restrict__ out,
                                            int batch) {
    __shared__ float Sre[4096];
    __shared__ float Sim[4096];
    __shared__ float pr[48];
    __shared__ float pi[48];
    __shared__ float evp[8][12];

    const int tid  = threadIdx.x;
    const int circ = blockIdx.x;
    const int g  = circ / batch;
    const int bb = circ - g * batch;
    const float* xb = x + bb * 192 + g * 12;   // 12 angles of this circuit

    // ---- per-block single-qubit product tables: p_b[j] = prod RX factors ----
    if (tid < 48) {
        const int blk = tid >> 4, j = tid & 15;
        float mag = 1.f; int k = 0;
        #pragma unroll
        for (int bit = 0; bit < 4; ++bit) {
            const int wire = 4*blk + (3 - bit);          // bit3 <-> wire 4b+0
            const float a = 0.5f * xb[wire];
            if ((j >> bit) & 1) { mag *= sinf(a); ++k; } // -i*sin amplitude
            else                {  mag *= cosf(a); }
        }
        float re, im;                                    // (-i)^k * mag
        switch (k & 3) {
            case 0:  re =  mag; im = 0.f;  break;
            case 1:  re = 0.f;  im = -mag; break;
            case 2:  re = -mag; im = 0.f;  break;
            default: re = 0.f;  im =  mag; break;
        }
        pr[tid] = re; pi[tid] = im;
    }
    __syncthreads();

    // ---- materialize initial product state ----
    {
        const int h = tid >> 4, m = tid & 15;
        const float ar = pr[h],    ai = pi[h];
        const float br = pr[16+m], bi = pi[16+m];
        const float cr = ar*br - ai*bi, ci = ar*bi + ai*br;
        const int base = tid * 16;
        #pragma unroll
        for (int l = 0; l < 16; ++l) {
            const float dr = pr[32+l], di = pi[32+l];
            Sre[base+l] = cr*dr - ci*di;
            Sim[base+l] = cr*di + ci*dr;
        }
    }
    __syncthreads();

    const int lane  = tid & 31;
    const int wv    = tid >> 5;
    const int nn    = lane & 15;          // A row / B,D column for this lane
    const int khalf = (lane >> 4) * 2;    // K sub-pair held by this half-wave
    const int dhalf = (lane >> 4) * 8;    // D row offset for this half-wave

    // ================= U2 on low index l:  Psi'[r,:] = Psi[r,:] * U2^T ======
    {
        const float* ure = ws + 1024;
        const float* uim = ws + 1280;
        v2f Br[4], Bi[4], Bn[4];
        #pragma unroll
        for (int kk = 0; kk < 4; ++kk) {
            const int k0 = 4*kk + khalf;
            Br[kk] = mk2(ure[nn*16 + k0], ure[nn*16 + k0 + 1]); // B[k,n]=U2[n,k]
            Bi[kk] = mk2(uim[nn*16 + k0], uim[nn*16 + k0 + 1]);
            Bn[kk] = -Bi[kk];
        }
        for (int t = wv*2; t < wv*2 + 2; ++t) {       // 16 row-tiles / 8 waves
            const int row = (t*16 + nn) * 16;
            v2f Ar[4], Ai[4];
            #pragma unroll
            for (int kk = 0; kk < 4; ++kk) {
                const int k0 = 4*kk + khalf;
                Ar[kk] = mk2(Sre[row + k0], Sre[row + k0 + 1]);
                Ai[kk] = mk2(Sim[row + k0], Sim[row + k0 + 1]);
            }
            v8f Dr = zero8(), Di = zero8();
            #pragma unroll
            for (int kk = 0; kk < 4; ++kk) Dr = wmma4(Ar[kk], Br[kk], Dr);
            #pragma unroll
            for (int kk = 0; kk < 4; ++kk) Dr = wmma4(Ai[kk], Bn[kk], Dr);
            #pragma unroll
            for (int kk = 0; kk < 4; ++kk) Di = wmma4(Ar[kk], Bi[kk], Di);
            #pragma unroll
            for (int kk = 0; kk < 4; ++kk) Di = wmma4(Ai[kk], Br[kk], Di);
            #pragma unroll
            for (int d = 0; d < 8; ++d) {
                const int o = (t*16 + d + dhalf)*16 + nn;
                Sre[o] = Dr[d]; Sim[o] = Di[d];
            }
        }
    }
    __syncthreads();

    // ================= U1 on middle index m:  per h-slab, Psi' = U1 * Psi ===
    {
        const float* ure = ws + 512;
        const float* uim = ws + 768;
        v2f Ar[4], Ai[4], An[4];
        #pragma unroll
        for (int kk = 0; kk < 4; ++kk) {
            const int k0 = 4*kk + khalf;
            Ar[kk] = mk2(ure[nn*16 + k0], ure[nn*16 + k0 + 1]); // A[m',k]=U1[m',k]
            Ai[kk] = mk2(uim[nn*16 + k0], uim[nn*16 + k0 + 1]);
            An[kk] = -Ai[kk];
        }
        for (int hh = wv*2; hh < wv*2 + 2; ++hh) {    // 16 slabs / 8 waves
            const int sb = hh * 256;
            v2f Br[4], Bi[4];
            #pragma unroll
            for (int kk = 0; kk < 4; ++kk) {
                const int k0 = 4*kk + khalf;
                Br[kk] = mk2(Sre[sb + k0*16 + nn], Sre[sb + (k0+1)*16 + nn]);
                Bi[kk] = mk2(Sim[sb + k0*16 + nn], Sim[sb + (k0+1)*16 + nn]);
            }
            v8f Dr = zero8(), Di = zero8();
            #pragma unroll
            for (int kk = 0; kk < 4; ++kk) Dr = wmma4(Ar[kk], Br[kk], Dr);
            #pragma unroll
            for (int kk = 0; kk < 4; ++kk) Dr = wmma4(An[kk], Bi[kk], Dr);
            #pragma unroll
            for (int kk = 0; kk < 4; ++kk) Di = wmma4(Ar[kk], Bi[kk], Di);
            #pragma unroll
            for (int kk = 0; kk < 4; ++kk) Di = wmma4(Ai[kk], Br[kk], Di);
            #pragma unroll
            for (int d = 0; d < 8; ++d) {
                const int o = sb + (d + dhalf)*16 + nn;
                Sre[o] = Dr[d]; Sim[o] = Di[d];
            }
        }
    }
    __syncthreads();

    // ================= U0 on high index h:  Psi' = U0 * Psi (col tiles) =====
    {
        const float* ure = ws + 0;
        const float* uim = ws + 256;
        v2f Ar[4], Ai[4], An[4];
        #pragma unroll
        for (int kk = 0; kk < 4; ++kk) {
            const int k0 = 4*kk + khalf;
            Ar[kk] = mk2(ure[nn*16 + k0], ure[nn*16 + k0 + 1]);
            Ai[kk] = mk2(uim[nn*16 + k0], uim[nn*16 + k0 + 1]);
            An[kk] = -Ai[kk];
        }
        for (int t = wv*2; t < wv*2 + 2; ++t) {       // 16 column tiles
            const int cb = t*16 + nn;
            v2f Br[4], Bi[4];
            #pragma unroll
            for (int kk = 0; kk < 4; ++kk) {
                const int k0 = 4*kk + khalf;
                Br[kk] = mk2(Sre[k0*256 + cb], Sre[(k0+1)*256 + cb]);
                Bi[kk] = mk2(Sim[k0*256 + cb], Sim[(k0+1)*256 + cb]);
            }
            v8f Dr = zero8(), Di = zero8();
            #pragma unroll
            for (int kk = 0; kk < 4; ++kk) Dr = wmma4(Ar[kk], Br[kk], Dr);
            #pragma unroll
            for (int kk = 0; kk < 4; ++kk) Dr = wmma4(An[kk], Bi[kk], Dr);
            #pragma unroll
            for (int kk = 0; kk < 4; ++kk) Di = wmma4(Ar[kk], Bi[kk], Di);
            #pragma unroll
            for (int kk = 0; kk < 4; ++kk) Di = wmma4(Ai[kk], Br[kk], Di);
            #pragma unroll
            for (int d = 0; d < 8; ++d) {
                const int o = (d + dhalf)*256 + cb;
                Sre[o] = Dr[d]; Sim[o] = Di[d];
            }
        }
    }
    __syncthreads();

    // ===== gate 12: control wire4 (i bit7), target wire8 (i bit3) ===========
    {
        const float t00r = ws[1536], t01r = ws[1537], t10r = ws[1538], t11r = ws[1539];
        const float t00i = ws[1540], t01i = ws[1541], t10i = ws[1542], t11i = ws[1543];
        #pragma unroll
        for (int q = 0; q < 4; ++q) {
            const int p  = tid*4 + q;                 // 1024 pairs
            const int h  = p >> 6, m2 = (p >> 3) & 7, l3 = p & 7;
            const int i0 = h*256 + (8 + m2)*16 + l3;  // ctrl=1, tgt=0
            const int i1 = i0 + 8;                    // tgt=1
            const float a0r = Sre[i0], a0i = Sim[i0];
            const float a1r = Sre[i1], a1i = Sim[i1];
            Sre[i0] = t00r*a0r - t00i*a0i + t01r*a1r - t01i*a1i;
            Sim[i0] = t00r*a0i + t00i*a0r + t01r*a1i + t01i*a1r;
            Sre[i1] = t10r*a0r - t10i*a0i + t11r*a1r - t11i*a1i;
            Sim[i1] = t10r*a0i + t10i*a0r + t11r*a1i + t11i*a1r;
        }
    }
    __syncthreads();

    // ===== gate 13: control wire0 (i bit11), target wire4 (i bit7) ==========
    {
        const float t00r = ws[1544], t01r = ws[1545], t10r = ws[1546], t11r = ws[1547];
        const float t00i = ws[1548], t01i = ws[1549], t10i = ws[1550], t11i = ws[1551];
        #pragma unroll
        for (int q = 0; q < 4; ++q) {
            const int p  = tid*4 + q;
            const int h2 = p >> 7, m3 = (p >> 4) & 7, l = p & 15;
            const int i0 = (8 + h2)*256 + m3*16 + l;
            const int i1 = i0 + 128;
            const float a0r = Sre[i0], a0i = Sim[i0];
            const float a1r = Sre[i1], a1i = Sim[i1];
            Sre[i0] = t00r*a0r - t00i*a0i + t01r*a1r - t01i*a1i;
            Sim[i0] = t00r*a0i + t00i*a0r + t01r*a1i + t01i*a1r;
            Sre[i1] = t10r*a0r - t10i*a0i + t11r*a1r - t11i*a1i;
            Sim[i1] = t10r*a0i + t10i*a0r + t11r*a1i + t11i*a1r;
        }
    }
    __syncthreads();

    // ===== <Z_w> expectations, deterministic reduction ======================
    float ev[12];
    #pragma unroll
    for (int w = 0; w < 12; ++w) ev[w] = 0.f;
    {
        const int base = tid * 16;
        for (int l = 0; l < 16; ++l) {
            const int i = base + l;
            const float pq = Sre[i]*Sre[i] + Sim[i]*Sim[i];
            #pragma unroll
            for (int w = 0; w < 12; ++w)
                ev[w] += ((i >> (11 - w)) & 1) ? -pq : pq;
        }
    }
    #pragma unroll
    for (int off = 16; off > 0; off >>= 1) {
        #pragma unroll
        for (int w = 0; w < 12; ++w) ev[w] += __shfl_xor(ev[w], off, 32);
    }
    if (lane == 0) {
        #pragma unroll
        for (int w = 0; w < 12; ++w) evp[wv][w] = ev[w];
    }
    __syncthreads();
    if (tid < 12) {
        float s = 0.f;
        #pragma unroll
        for (int v = 0; v < 8; ++v) s += evp[v][tid];   // fixed order
        out[circ*12 + tid] = s;
    }
}

extern "C" void kernel_launch(void* const* d_in, const int* in_sizes, int n_in,
                              void* d_out, int out_size, void* d_ws, size_t ws_size,
                              hipStream_t stream) {
    (void)n_in; (void)out_size; (void)ws_size;
    const float* x = (const float*)d_in[0];        // [batch, 48, 2, 2] f32
    const float* W = (const float*)d_in[1];        // [28] f32
    float* ws  = (float*)d_ws;                     // 1552 floats used
    float* out = (float*)d_out;                    // [batch*48*4] f32

    const int batch    = in_sizes[0] / 192;        // 256
    const int circuits = batch * 16;               // 4096

    build_unitaries<<<3, 256, 0, stream>>>(W, ws);
    qsim<<<circuits, 256, 0, stream>>>(x, ws, out, batch);
}